// PretextGenerator_73933567034026
// MI455X (gfx1250) — compile-verified
//
#include <hip/hip_runtime.h>
#include <stdint.h>

// VIME pretext generator for MI455X (gfx1250).
// Pure data-movement op: double-buffered async-DMA streaming (ASYNCcnt path),
// L2-resident scattered gather, wave32-sized coalesced b128 traffic.
// Permutation indices are software-pipelined: computed once per element,
// used for prefetch at t, reused for the gather at t+1.

#define M_ROWS        131072      // 2^17 rows
#define N_COLS        256
#define TPB           256         // 8 wave32 waves
#define ROWS_PER_TILE 4           // rows per block per pipeline stage
#define GRID_BLOCKS   2048
#define ROW_STRIDE    (GRID_BLOCKS * ROWS_PER_TILE)   // 8192; 131072/8192 = 16 iters

#define AS1 __attribute__((address_space(1)))
#define AS3 __attribute__((address_space(3)))

typedef int v4i __attribute__((vector_size(16)));

#if __has_builtin(__builtin_amdgcn_s_wait_asynccnt)
#define WAIT_ASYNC(n) do { __builtin_amdgcn_s_wait_asynccnt(n); \
                           asm volatile("" ::: "memory"); } while (0)
#else
#define WAIT_ASYNC(n) asm volatile("s_wait_asynccnt " #n ::: "memory")
#endif

// Async global -> LDS copy of 16 bytes per lane (GLOBAL_LOAD_ASYNC_TO_LDS_B128,
// tracked by ASYNCcnt).
__device__ __forceinline__ void async_copy16(const float* __restrict__ base,
                                             uint32_t byteOff, void* ldsDst) {
#if __has_builtin(__builtin_amdgcn_global_load_async_to_lds_b128)
  __builtin_amdgcn_global_load_async_to_lds_b128(
      (AS1 v4i*)(uintptr_t)((const char*)base + byteOff),
      (AS3 v4i*)(uint32_t)(uintptr_t)ldsDst,
      /*imm offset*/ 0, /*cpol*/ 0);
#else
  uint32_t ldsOff = (uint32_t)(uintptr_t)ldsDst;   // low 32 bits = LDS byte offset
  asm volatile("global_load_async_to_lds_b128 %0, %1, %2"
               :
               : "v"(ldsOff), "v"(byteOff), "s"(base)
               : "memory");
#endif
}

// Per-column key (seeded with 42, mirroring the reference's perm_key intent).
__device__ __forceinline__ uint32_t colkey(uint32_t j) {
  uint32_t h = j * 0x9E3779B9u + 42u * 0x85EBCA6Bu;
  h ^= h >> 16; h *= 0x7FEB352Du;
  h ^= h >> 15; h *= 0x846CA68Bu;
  h ^= h >> 16;
  return h;
}

// Keyed bijection on [0, 2^17): xor-const, odd-multiply mod 2^17, xorshift-down,
// odd-multiply mod 2^17 — each step invertible, so a true row permutation,
// independently keyed per column. 4 steps (~6 VALU ops) per element.
__device__ __forceinline__ uint32_t perm17(uint32_t i, uint32_t key) {
  uint32_t p = (i ^ key) & 0x1FFFFu;
  p = (p * ((key >> 8) | 1u)) & 0x1FFFFu;
  p ^= p >> 8;
  p = (p * 0x6C79u) & 0x1FFFFu;
  return p;
}

__global__ __launch_bounds__(TPB, 4)
void vime_pretext_kernel(const float* __restrict__ x,
                         const float* __restrict__ mask,
                         float* __restrict__ out_cx,
                         float* __restrict__ out_cm) {
  // Double-buffered staging tiles: 4 rows x 256 cols x 4B = 4KB each, x2 streams, x2 stages.
  __shared__ __align__(16) float s_x[2][TPB * 4];
  __shared__ __align__(16) float s_m[2][TPB * 4];

  const int tid     = threadIdx.x;
  const int colLane = tid & 63;          // 64 float4-groups span the 256 columns
  const int rowLane = tid >> 6;          // 4 rows per tile
  const int col0    = colLane << 2;
  const int ldsIdx  = tid << 2;          // 16B per thread, private slot

  uint32_t key[4];
#pragma unroll
  for (int i = 0; i < 4; ++i) key[i] = colkey((uint32_t)(col0 + i));

  int row = (int)blockIdx.x * ROWS_PER_TILE + rowLane;
  int buf = 0;

  // Prologue: issue async stage 0 and compute stage-0 gather indices.
  uint32_t pcur[4];
  {
    uint32_t off = ((uint32_t)row * N_COLS + (uint32_t)col0) * 4u;
    async_copy16(x,    off, &s_x[0][ldsIdx]);
    async_copy16(mask, off, &s_m[0][ldsIdx]);
#pragma unroll
    for (int i = 0; i < 4; ++i) pcur[i] = perm17((uint32_t)row, key[i]);
  }

  while (true) {
    const int  nextRow = row + ROW_STRIDE;
    const bool more    = (nextRow < M_ROWS);   // uniform per wave (rowLane-invariant stride)

    uint32_t pnext[4];
    if (more) {
      // Issue stage t+1 while stage t drains.
      uint32_t off = ((uint32_t)nextRow * N_COLS + (uint32_t)col0) * 4u;
      async_copy16(x,    off, &s_x[buf ^ 1][ldsIdx]);
      async_copy16(mask, off, &s_m[buf ^ 1][ldsIdx]);
      // Compute t+1 gather indices once; warm L0 for their (L2-resident) lines.
#pragma unroll
      for (int i = 0; i < 4; ++i) {
        pnext[i] = perm17((uint32_t)nextRow, key[i]);
        __builtin_prefetch((const void*)(x + (((size_t)pnext[i]) << 8) + (size_t)(col0 + i)), 0, 1);
      }
      WAIT_ASYNC(2);   // async ops complete in order: stage t's 2 ops are done
    } else {
      WAIT_ASYNC(0);
    }

    const float4 xv = *(const float4*)&s_x[buf][ldsIdx];
    const float4 mv = *(const float4*)&s_m[buf][ldsIdx];

    float sh[4];
#pragma unroll
    for (int i = 0; i < 4; ++i)
      sh[i] = x[(((size_t)pcur[i]) << 8) + (size_t)(col0 + i)];   // gather within column

    const float c0 = xv.x * (1.0f - mv.x) + sh[0] * mv.x;
    const float c1 = xv.y * (1.0f - mv.y) + sh[1] * mv.y;
    const float c2 = xv.z * (1.0f - mv.z) + sh[2] * mv.z;
    const float c3 = xv.w * (1.0f - mv.w) + sh[3] * mv.w;

    float4 cxv = make_float4(c0, c1, c2, c3);
    float4 cmv = make_float4((xv.x != c0) ? 1.0f : 0.0f,
                             (xv.y != c1) ? 1.0f : 0.0f,
                             (xv.z != c2) ? 1.0f : 0.0f,
                             (xv.w != c3) ? 1.0f : 0.0f);

    const size_t outOff = (size_t)row * N_COLS + (size_t)col0;
    *(float4*)(out_cx + outOff) = cxv;   // coalesced b128 stores
    *(float4*)(out_cm + outOff) = cmv;

    if (!more) break;
#pragma unroll
    for (int i = 0; i < 4; ++i) pcur[i] = pnext[i];   // carry pipelined indices
    row = nextRow;
    buf ^= 1;
  }
}

extern "C" void kernel_launch(void* const* d_in, const int* in_sizes, int n_in,
                              void* d_out, int out_size, void* d_ws, size_t ws_size,
                              hipStream_t stream) {
  (void)in_sizes; (void)n_in; (void)out_size; (void)d_ws; (void)ws_size;
  const float* x    = (const float*)d_in[0];
  const float* mask = (const float*)d_in[1];
  float* out_cx = (float*)d_out;                              // corrupt_x  [M*N]
  float* out_cm = out_cx + (size_t)M_ROWS * (size_t)N_COLS;   // corrupt_mask [M*N]
  vime_pretext_kernel<<<GRID_BLOCKS, TPB, 0, stream>>>(x, mask, out_cx, out_cm);
}